// myGATConv_19026705122036
// MI455X (gfx1250) — compile-verified
//
#include <hip/hip_runtime.h>
#include <hip/hip_bf16.h>

typedef __attribute__((ext_vector_type(2))) float v2f;
typedef __attribute__((ext_vector_type(8))) float v8f;

#define GN 50000
#define GE 800000
#define GH 4
#define GD 64
#define GIN 256
#define GET 8
#define SLOPE 0.2f

// ---------------------------------------------------------------------------
// order-preserving float<->uint key for atomic segment-max
// ---------------------------------------------------------------------------
__device__ __forceinline__ unsigned fkey(float x) {
    unsigned u = __float_as_uint(x);
    return (u & 0x80000000u) ? ~u : (u | 0x80000000u);
}
__device__ __forceinline__ float funkey(unsigned k) {
    return (k & 0x80000000u) ? __uint_as_float(k ^ 0x80000000u)
                             : __uint_as_float(~k);
}

// ---------------------------------------------------------------------------
// K0: init accumulators: rstacc = 0, denom = 0, emax key = key(-inf)
// ---------------------------------------------------------------------------
__global__ void init_ws_kernel(float* __restrict__ rstacc,
                               float* __restrict__ denom,
                               unsigned* __restrict__ emaxu,
                               int nAcc, int nNH) {
    int i = blockIdx.x * blockDim.x + threadIdx.x;
    if (i < nAcc) rstacc[i] = 0.0f;
    if (i < nNH) {
        denom[i] = 0.0f;
        emaxu[i] = 0x007FFFFFu; // fkey(-inf)
    }
}

// ---------------------------------------------------------------------------
// K1: dual projection GEMM with V_WMMA_F32_16X16X4_F32
//     out[m][n] = sum_k feat[m][k] * W[n][k]       (W row-major [256][256])
//     blockIdx.y: 0 -> W_fc/ft, 1 -> W_res/res
//     block = 256 thr (8 waves); block tile 16(M) x 256(N); wave tile 16x32
// ---------------------------------------------------------------------------
__global__ __launch_bounds__(256) void proj_gemm_kernel(
        const float* __restrict__ feat,
        const float* __restrict__ Wfc,
        const float* __restrict__ Wres,
        float* __restrict__ ft,
        float* __restrict__ res) {
    __shared__ float As[16 * GIN]; // 16 KB activation tile

    const int m0  = blockIdx.x * 16;
    const float* W   = (blockIdx.y == 0) ? Wfc : Wres;
    float*       out = (blockIdx.y == 0) ? ft  : res;

    const int tid = threadIdx.x;
    // stage 16x256 fp32 tile of feat into LDS (1024 float4, 4 per thread)
    {
        const float4* s4 = reinterpret_cast<const float4*>(feat + (size_t)m0 * GIN);
        float4* d4 = reinterpret_cast<float4*>(As);
#pragma unroll
        for (int i = 0; i < 4; ++i) d4[tid + i * 256] = s4[tid + i * 256];
    }
    __syncthreads();

    const int wave  = tid >> 5;
    const int lane  = tid & 31;
    const int n0    = wave * 32;          // wave's N-tile base (two 16-wide tiles)
    const int mrow  = lane & 15;          // A: M index / B: N index
    const int khalf = (lane >> 4) * 2;    // lanes 16-31 hold K+2,K+3

    v8f acc0 = {};
    v8f acc1 = {};

    const float* Wr0 = W + (size_t)(n0 + mrow) * GIN;
    const float* Wr1 = W + (size_t)(n0 + 16 + mrow) * GIN;
    const float* Arow = As + mrow * GIN;

    for (int k0 = 0; k0 < GIN; k0 += 4) {
        v2f a  = *reinterpret_cast<const v2f*>(Arow + k0 + khalf);
        v2f b0 = *reinterpret_cast<const v2f*>(Wr0 + k0 + khalf);
        v2f b1 = *reinterpret_cast<const v2f*>(Wr1 + k0 + khalf);
        acc0 = __builtin_amdgcn_wmma_f32_16x16x4_f32(
            false, a, false, b0, (short)0, acc0, false, false);
        acc1 = __builtin_amdgcn_wmma_f32_16x16x4_f32(
            false, a, false, b1, (short)0, acc1, false, false);
    }

    // C/D layout: VGPR v -> row v + 8*(lane>=16), col = lane&15
    const int crow = (lane >> 4) * 8;
    const int ccol = lane & 15;
#pragma unroll
    for (int v = 0; v < 8; ++v) {
        size_t r = (size_t)(m0 + crow + v) * GIN;
        out[r + n0 + ccol]      = acc0[v];
        out[r + n0 + 16 + ccol] = acc1[v];
    }
}

// ---------------------------------------------------------------------------
// K2: ee[t][h] = sum_f attn_e[h][f] * (edge_emb[t] . W_e[h*64+f])
//     tiny: 8 types x 4 heads; one thread per (t,h)
// ---------------------------------------------------------------------------
__global__ void ee_kernel(const float* __restrict__ edge_emb,
                          const float* __restrict__ We,
                          const float* __restrict__ attn_e,
                          float* __restrict__ ee) {
    int i = threadIdx.x;
    if (i >= GET * GH) return;
    int t = i >> 2, h = i & 3;
    const float* erow = edge_emb + t * 64;
    float s = 0.0f;
    for (int f = 0; f < 64; ++f) {
        const float* wrow = We + (size_t)(h * 64 + f) * 64;
        float d = 0.0f;
#pragma unroll 4
        for (int k = 0; k < 64; ++k) d += erow[k] * wrow[k];
        s += d * attn_e[h * 64 + f];
    }
    ee[t * GH + h] = s;
}

// ---------------------------------------------------------------------------
// K3: el[n][h] = ft[n,h,:].attn_l[h]  ;  er likewise
// ---------------------------------------------------------------------------
__global__ void elr_kernel(const float* __restrict__ ft,
                           const float* __restrict__ attn_l,
                           const float* __restrict__ attn_r,
                           float* __restrict__ el,
                           float* __restrict__ er) {
    int i = blockIdx.x * blockDim.x + threadIdx.x;
    if (i >= GN * GH) return;
    int n = i >> 2, h = i & 3;
    const float* row = ft + (size_t)n * (GH * GD) + h * GD;
    const float* al = attn_l + h * GD;
    const float* ar = attn_r + h * GD;
    float sl = 0.0f, sr = 0.0f;
#pragma unroll 4
    for (int d = 0; d < GD; ++d) {
        float v = row[d];
        sl += v * al[d];
        sr += v * ar[d];
    }
    el[i] = sl;
    er[i] = sr;
}

// ---------------------------------------------------------------------------
// K4: edge logits + leaky relu + atomic segment-max over dst
// ---------------------------------------------------------------------------
__global__ void logits_kernel(const int* __restrict__ src,
                              const int* __restrict__ dst,
                              const int* __restrict__ etype,
                              const float* __restrict__ el,
                              const float* __restrict__ er,
                              const float* __restrict__ ee,
                              float* __restrict__ ebuf,
                              unsigned* __restrict__ emaxu) {
    int i = blockIdx.x * blockDim.x + threadIdx.x;
    if (i >= GE * GH) return;
    int eid = i >> 2, h = i & 3;
    int s = src[eid], d = dst[eid], t = etype[eid];
    float v = el[s * GH + h] + er[d * GH + h] + ee[t * GH + h];
    v = (v > 0.0f) ? v : SLOPE * v;
    ebuf[i] = v;
    atomicMax(&emaxu[d * GH + h], fkey(v));
}

// ---------------------------------------------------------------------------
// K5: z = exp(e - emax[dst]); denom += z   (z overwrites ebuf)
// ---------------------------------------------------------------------------
__global__ void expsum_kernel(const int* __restrict__ dst,
                              float* __restrict__ ebuf,
                              const unsigned* __restrict__ emaxu,
                              float* __restrict__ denom) {
    int i = blockIdx.x * blockDim.x + threadIdx.x;
    if (i >= GE * GH) return;
    int eid = i >> 2, h = i & 3;
    int d = dst[eid];
    float m = funkey(emaxu[d * GH + h]);
    if (!(m > -3.0e38f)) m = 0.0f; // guard -inf (0-in-degree), mirrors reference
    float z = __expf(ebuf[i] - m);
    ebuf[i] = z;
    atomicAdd(&denom[d * GH + h], z);
}

// ---------------------------------------------------------------------------
// K6: a = z / max(denom[dst], 1e-9)  -> second output region
// ---------------------------------------------------------------------------
__global__ void norm_kernel(const int* __restrict__ dst,
                            const float* __restrict__ zbuf,
                            const float* __restrict__ denom,
                            float* __restrict__ aout) {
    int i = blockIdx.x * blockDim.x + threadIdx.x;
    if (i >= GE * GH) return;
    int eid = i >> 2, h = i & 3;
    aout[i] = zbuf[i] / fmaxf(denom[dst[eid] * GH + h], 1e-9f);
}

// ---------------------------------------------------------------------------
// K7: aggregation: rstacc[dst,h,:] += a[e,h] * ft[src,h,:]
//     one wave per (edge, head); lane covers 2 contiguous floats (float2)
// ---------------------------------------------------------------------------
__global__ __launch_bounds__(256) void aggregate_kernel(
        const int* __restrict__ src,
        const int* __restrict__ dst,
        const float* __restrict__ aout,
        const float* __restrict__ ft,
        float* __restrict__ rstacc) {
    int task = blockIdx.x * 8 + (threadIdx.x >> 5);
    if (task >= GE * GH) return;
    int lane = threadIdx.x & 31;
    int eid = task >> 2, h = task & 3;
    float coef = aout[task];
    int s = src[eid], d = dst[eid];
    const float2 v = *reinterpret_cast<const float2*>(
        ft + (size_t)s * (GH * GD) + h * GD + lane * 2);
    float* o = rstacc + (size_t)d * (GH * GD) + h * GD + lane * 2;
    atomicAdd(o,     v.x * coef);
    atomicAdd(o + 1, v.y * coef);
}

// ---------------------------------------------------------------------------
// K8: rst = ELU(acc + res + bias)
// ---------------------------------------------------------------------------
__global__ void finalize_kernel(const float* __restrict__ rstacc,
                                const float* __restrict__ res,
                                const float* __restrict__ bias,
                                float* __restrict__ outr) {
    int i = blockIdx.x * blockDim.x + threadIdx.x;
    if (i >= GN * GH * GD) return;
    float x = rstacc[i] + res[i] + bias[i & (GH * GD - 1)];
    outr[i] = (x > 0.0f) ? x : (__expf(x) - 1.0f);
}

// ---------------------------------------------------------------------------
extern "C" void kernel_launch(void* const* d_in, const int* in_sizes, int n_in,
                              void* d_out, int out_size, void* d_ws, size_t ws_size,
                              hipStream_t stream) {
    const float* feat     = (const float*)d_in[0];
    const int*   etype    = (const int*)  d_in[1];
    const int*   src      = (const int*)  d_in[2];
    const int*   dst      = (const int*)  d_in[3];
    const float* Wfc      = (const float*)d_in[4];
    const float* We       = (const float*)d_in[5];
    const float* edge_emb = (const float*)d_in[6];
    const float* attn_l   = (const float*)d_in[7];
    const float* attn_r   = (const float*)d_in[8];
    const float* attn_e   = (const float*)d_in[9];
    const float* Wres     = (const float*)d_in[10];
    const float* bias     = (const float*)d_in[11];

    (void)in_sizes; (void)n_in; (void)out_size; (void)ws_size;

    const size_t NC  = (size_t)GN * GH * GD;   // 12.8M  node feature block
    const size_t NH  = (size_t)GN * GH;        // 200K
    const size_t EH  = (size_t)GE * GH;        // 3.2M

    float* ws      = (float*)d_ws;
    float* ft      = ws;                       // [N,256]
    float* res     = ft + NC;                  // [N,256]
    float* rstacc  = res + NC;                 // [N,256]
    float* el      = rstacc + NC;              // [N,H]
    float* er      = el + NH;                  // [N,H]
    float* ee      = er + NH;                  // [ET,H]
    float* ebuf    = ee + (size_t)GET * GH;    // [E,H] logits, then z
    float* denom   = ebuf + EH;                // [N,H]
    unsigned* emaxu = (unsigned*)(denom + NH); // [N,H]

    float* outr = (float*)d_out;               // [N,H,D]
    float* aout = outr + NC;                   // [E,H]

    // K0: init
    {
        int n = (int)NC;
        init_ws_kernel<<<(n + 255) / 256, 256, 0, stream>>>(
            rstacc, denom, emaxu, n, (int)NH);
    }
    // K1: dual GEMM (ft, res) via WMMA f32 16x16x4
    {
        dim3 grid(GN / 16, 2);
        proj_gemm_kernel<<<grid, 256, 0, stream>>>(feat, Wfc, Wres, ft, res);
    }
    // K2: ee table
    ee_kernel<<<1, 32, 0, stream>>>(edge_emb, We, attn_e, ee);
    // K3: el / er
    elr_kernel<<<((int)NH + 255) / 256, 256, 0, stream>>>(ft, attn_l, attn_r, el, er);
    // K4: logits + leaky relu + segment max
    logits_kernel<<<((int)EH + 255) / 256, 256, 0, stream>>>(
        src, dst, etype, el, er, ee, ebuf, emaxu);
    // K5: exp + denom
    expsum_kernel<<<((int)EH + 255) / 256, 256, 0, stream>>>(dst, ebuf, emaxu, denom);
    // K6: normalize -> a output
    norm_kernel<<<((int)EH + 255) / 256, 256, 0, stream>>>(dst, ebuf, denom, aout);
    // K7: weighted aggregation (wave per edge-head)
    aggregate_kernel<<<((int)EH + 7) / 8, 256, 0, stream>>>(src, dst, aout, ft, rstacc);
    // K8: residual + bias + ELU -> rst output
    finalize_kernel<<<((int)NC + 255) / 256, 256, 0, stream>>>(rstacc, res, bias, outr);
}